// DetectionLoss_40535901339910
// MI455X (gfx1250) — compile-verified
//
#include <hip/hip_runtime.h>
#include <math.h>

#define BI 16
#define AI 65536
#define GI 32
#define NB 64            // partial blocks per image for match/negsum
#define CHUNK (AI / NB)  // 1024 anchors per partial block
#define IOU_TH 0.5f

#if defined(__has_builtin)
#  if __has_builtin(__builtin_amdgcn_global_load_async_to_lds_b32) && __has_builtin(__builtin_amdgcn_s_wait_asynccnt)
#    define USE_ASYNC_LDS 1
#  endif
#endif
#ifndef USE_ASYNC_LDS
#  define USE_ASYNC_LDS 0
#endif

#if USE_ASYNC_LDS
typedef __attribute__((address_space(1))) int* as1_int_ptr;
typedef __attribute__((address_space(3))) int* as3_int_ptr;
#endif

// ---------------------------------------------------------------- helpers
__device__ __forceinline__ float smoothL1(float d) {
  float ad = fabsf(d);
  return ad < 1.0f ? 0.5f * ad * ad : ad - 0.5f;
}

__device__ __forceinline__ float blockReduceF(float v, float* sred) {
  sred[threadIdx.x] = v;
  __syncthreads();
  for (int s = 128; s > 0; s >>= 1) {
    if (threadIdx.x < (unsigned)s) sred[threadIdx.x] += sred[threadIdx.x + s];
    __syncthreads();
  }
  float r = sred[0];
  __syncthreads();
  return r;
}

__device__ __forceinline__ int blockReduceI(int v, int* ired) {
  ired[threadIdx.x] = v;
  __syncthreads();
  for (int s = 128; s > 0; s >>= 1) {
    if (threadIdx.x < (unsigned)s) ired[threadIdx.x] += ired[threadIdx.x + s];
    __syncthreads();
  }
  int r = ired[0];
  __syncthreads();
  return r;
}

// ---------------------------------------------------------------- K0: init
__global__ void k_init(unsigned long long* bestAnchor, unsigned* hist) {
  for (int j = threadIdx.x; j < BI * GI; j += blockDim.x) bestAnchor[j] = 0ull;
  for (int j = threadIdx.x; j < BI * 256; j += blockDim.x) hist[j] = 0u;
}

// ---------------------------------------------------------------- K1: IoU + argmaxes
__global__ __launch_bounds__(256) void k_iou(
    const float4* __restrict__ anchors, const float4* __restrict__ gt,
    float* __restrict__ bestIou, int* __restrict__ bestIdx,
    unsigned long long* __restrict__ bestAnchor) {
  const int b = blockIdx.y;
  const int a = blockIdx.x * 256 + threadIdx.x;

  __shared__ float4 sgt[GI];
  __shared__ unsigned long long skey[GI];
  if (threadIdx.x < GI) skey[threadIdx.x] = 0ull;

#if USE_ASYNC_LDS
  if (threadIdx.x < GI * 4) {
    const float* src = ((const float*)gt) + (size_t)b * GI * 4 + threadIdx.x;
    float* dst = ((float*)sgt) + threadIdx.x;
    // gfx1250 async DMA path: global_load_async_to_lds_b32 (ASYNCcnt)
    __builtin_amdgcn_global_load_async_to_lds_b32(
        (as1_int_ptr)(unsigned long long)(uintptr_t)src,
        (as3_int_ptr)(unsigned)(uintptr_t)dst, 0, 0);
    __builtin_amdgcn_s_wait_asynccnt(0);
  }
#else
  if (threadIdx.x < GI * 4) {
    ((float*)sgt)[threadIdx.x] = ((const float*)gt)[(size_t)b * GI * 4 + threadIdx.x];
  }
#endif
  __syncthreads();

  // gfx1250 global_prefetch_b8: warm next anchor tile
  __builtin_prefetch(&anchors[(a + 4096) & (AI - 1)], 0, 1);

  float4 an = anchors[a];
  float area_a = (an.z - an.x) * (an.w - an.y);
  float best = -1.0f;
  int bidx = 0;
#pragma unroll 4
  for (int g = 0; g < GI; ++g) {
    float4 gb = sgt[g];
    float lx = fmaxf(an.x, gb.x), ly = fmaxf(an.y, gb.y);
    float rx = fminf(an.z, gb.z), ry = fminf(an.w, gb.w);
    float w = fmaxf(rx - lx, 0.0f), h = fmaxf(ry - ly, 0.0f);
    float inter = w * h;
    float area_g = (gb.z - gb.x) * (gb.w - gb.y);
    float iou = inter / (area_a + area_g - inter + 1e-6f);
    if (iou > best) { best = iou; bidx = g; }  // first-index argmax over G
    // per-gt argmax over A: larger iou wins; tie -> smaller anchor idx wins.
    // Racy pre-read guard: max is monotone, so skipping when key <= snapshot
    // is exact; avoids ~8K serialized same-address LDS atomics per block.
    unsigned long long key =
        ((unsigned long long)__float_as_uint(iou) << 32) |
        (unsigned long long)(0xFFFFFFFFu - (unsigned)a);
    if (key > skey[g]) atomicMax(&skey[g], key);
  }
  bestIou[(size_t)b * AI + a] = best;
  bestIdx[(size_t)b * AI + a] = bidx;
  __syncthreads();
  if (threadIdx.x < GI)
    atomicMax(&bestAnchor[b * GI + threadIdx.x], skey[threadIdx.x]);
}

// ---------------------------------------------------------------- K2: match / BCE / loc (partials)
__global__ __launch_bounds__(256) void k_match_p(
    const float4* __restrict__ bbox, const float* __restrict__ conf,
    const float4* __restrict__ gt, const float* __restrict__ bestIou,
    const int* __restrict__ bestIdx,
    const unsigned long long* __restrict__ bestAnchor,
    unsigned* __restrict__ negU, int* __restrict__ npP, int* __restrict__ pcP,
    float* __restrict__ llP, float* __restrict__ pbP, float* __restrict__ isP,
    int* __restrict__ icP) {
  const int b = blockIdx.y;
  const int base = blockIdx.x * CHUNK;
  __shared__ float4 sgt[GI];
  __shared__ int sba[GI];
  __shared__ float sred[256];
  __shared__ int ired[256];
  if (threadIdx.x < GI) {
    sgt[threadIdx.x] = gt[b * GI + threadIdx.x];
    sba[threadIdx.x] =
        (int)(0xFFFFFFFFu -
              (unsigned)(bestAnchor[b * GI + threadIdx.x] & 0xFFFFFFFFull));
  }
  __syncthreads();

  int r[GI];  // register-resident best-anchor ids
#pragma unroll
  for (int g = 0; g < GI; ++g) r[g] = sba[g];

  int np = 0, pc = 0, icnt = 0;
  float ll = 0.0f, pb = 0.0f, isum = 0.0f;
  for (int i = threadIdx.x; i < CHUNK; i += 256) {
    const int a = base + i;
    float bi = bestIou[(size_t)b * AI + a];
    bool pos = bi > IOU_TH;
#pragma unroll
    for (int g = 0; g < GI; ++g) pos = pos || (r[g] == a);
    float p = conf[(size_t)b * AI + a];
    float logp = fmaxf(logf(p), -100.0f);
    float log1mp = fmaxf(log1pf(-p), -100.0f);
    if (pos) {
      np += 1;
      pb += -logp;  // BCE target 1
      float4 pr = bbox[(size_t)b * AI + a];
      float4 mt = sgt[bestIdx[(size_t)b * AI + a]];
      float d0 = (pr.x + pr.z) * 0.5f - (mt.x + mt.z) * 0.5f;
      float d1 = (pr.y + pr.w) * 0.5f - (mt.y + mt.w) * 0.5f;
      float d2 = (pr.z - pr.x) - (mt.z - mt.x);
      float d3 = (pr.w - pr.y) - (mt.w - mt.y);
      ll += smoothL1(d0) + smoothL1(d1) + smoothL1(d2) + smoothL1(d3);
      negU[(size_t)b * AI + a] = 0u;
    } else {
      negU[(size_t)b * AI + a] = __float_as_uint(-log1mp);  // BCE target 0, > 0
    }
    pc += (p > 0.5f) ? 1 : 0;
    if (b == BI - 1 && bi > 0.0f) { isum += bi; icnt += 1; }
  }
  int npT = blockReduceI(np, ired);
  int pcT = blockReduceI(pc, ired);
  float llT = blockReduceF(ll, sred);
  float pbT = blockReduceF(pb, sred);
  float isT = blockReduceF(isum, sred);
  int icT = blockReduceI(icnt, ired);
  if (threadIdx.x == 0) {
    const int part = b * NB + blockIdx.x;
    npP[part] = npT;
    pcP[part] = pcT;
    llP[part] = llT;
    pbP[part] = pbT;
    isP[part] = isT;
    icP[part] = icT;
  }
}

// ---------------------------------------------------------------- K3: combine match partials (fixed order)
__global__ void k_combine(const int* npP, const int* pcP, const float* llP,
                          const float* pbP, const float* isP, const int* icP,
                          int* numPos, int* predCnt, float* locL, float* posBce,
                          int* kNeg, int* kRem, unsigned* prefix, float* iouSum,
                          int* iouCnt) {
  int b = threadIdx.x;
  if (b >= BI) return;
  int np = 0, pc = 0, ic = 0;
  float ll = 0.0f, pb = 0.0f, is = 0.0f;
  for (int j = 0; j < NB; ++j) {
    np += npP[b * NB + j];
    pc += pcP[b * NB + j];
    ll += llP[b * NB + j];
    pb += pbP[b * NB + j];
    is += isP[b * NB + j];
    ic += icP[b * NB + j];
  }
  numPos[b] = np;
  predCnt[b] = pc;
  locL[b] = ll;
  posBce[b] = pb;
  int k = min(3 * np, AI - np);
  kNeg[b] = k;
  kRem[b] = k;
  prefix[b] = 0u;
  if (b == BI - 1) { *iouSum = is; *iouCnt = ic; }
}

// ---------------------------------------------------------------- K4: radix-select histogram pass
__global__ __launch_bounds__(256) void k_hist(
    const unsigned* __restrict__ negU, const unsigned* __restrict__ prefix,
    unsigned* __restrict__ hist, int digit) {
  const int b = blockIdx.y;
  __shared__ unsigned sh[256];
  sh[threadIdx.x] = 0u;
  __syncthreads();
  unsigned pre = prefix[b];
  int shift = 8 * digit;
  for (int a = blockIdx.x * 256 + threadIdx.x; a < AI; a += gridDim.x * 256) {
    unsigned u = negU[(size_t)b * AI + a];
    bool ok = (digit == 3) || ((u >> (shift + 8)) == pre);
    if (ok) atomicAdd(&sh[(u >> shift) & 0xFFu], 1u);
  }
  __syncthreads();
  unsigned c = sh[threadIdx.x];
  if (c) atomicAdd(&hist[b * 256 + threadIdx.x], c);
}

// ---------------------------------------------------------------- K5: radix-select scan (tiny)
__global__ void k_scan(unsigned* hist, unsigned* prefix, int* kRem) {
  int b = threadIdx.x;
  if (b < BI) {
    int k = kRem[b];
    if (k > 0) {
      unsigned pre = prefix[b];
      unsigned cum = 0;
      int sel = -1;
      for (int bin = 255; bin >= 0 && sel < 0; --bin) {
        unsigned c = hist[b * 256 + bin];
        if (cum + c >= (unsigned)k) sel = bin;
        else cum += c;
      }
      if (sel < 0) sel = 0;
      prefix[b] = (pre << 8) | (unsigned)sel;
      kRem[b] = k - (int)cum;
    }
    for (int bin = 0; bin < 256; ++bin) hist[b * 256 + bin] = 0u;  // reset
  }
}

// ---------------------------------------------------------------- K6: top-k negative sum (partials)
__global__ __launch_bounds__(256) void k_negsum_p(
    const unsigned* __restrict__ negU, const unsigned* __restrict__ prefix,
    float* __restrict__ nsP, int* __restrict__ ncP) {
  const int b = blockIdx.y;
  const int base = blockIdx.x * CHUNK;
  __shared__ float sred[256];
  __shared__ int ired[256];
  unsigned vk = prefix[b];
  float s = 0.0f;
  int c = 0;
  for (int i = threadIdx.x; i < CHUNK; i += 256) {
    unsigned u = negU[(size_t)b * AI + base + i];
    if (u > vk) { s += __uint_as_float(u); c++; }
  }
  float sT = blockReduceF(s, sred);
  int cT = blockReduceI(c, ired);
  if (threadIdx.x == 0) {
    nsP[b * NB + blockIdx.x] = sT;
    ncP[b * NB + blockIdx.x] = cT;
  }
}

// ---------------------------------------------------------------- K7: finalize
__global__ void k_final(const int* numPos, const int* predCnt,
                        const float* locL, const float* posBce,
                        const float* nsP, const int* ncP,
                        const unsigned* prefix, const int* kNeg,
                        const float* iouSum, const int* iouCnt, float* out) {
  __shared__ float sNeg[BI];
  int b = threadIdx.x;
  if (b < BI) {
    float s = 0.0f;
    int c = 0;
    for (int j = 0; j < NB; ++j) { s += nsP[b * NB + j]; c += ncP[b * NB + j]; }
    int k = kNeg[b];
    unsigned vk = prefix[b];
    sNeg[b] = (k > 0) ? (s + (float)(k - c) * __uint_as_float(vk)) : 0.0f;
  }
  __syncthreads();
  if (threadIdx.x != 0) return;
  int npT = 0;
  float loc = 0.0f, confl = 0.0f, cl = 0.0f;
  for (int b2 = 0; b2 < BI; ++b2) {
    npT += numPos[b2];
    loc += locL[b2];
    confl += posBce[b2] + sNeg[b2];
  }
  for (int b2 = 0; b2 < BI; ++b2) {
    float diff = fabsf((float)predCnt[b2] - (float)GI);
    float pen = (diff <= 3.0f) ? diff * 0.2f
                               : 0.6f + 0.2f * logf(fmaxf(diff - 2.0f, 1.0f));
    cl += pen;
  }
  cl /= (float)BI;
  float npos = fmaxf(1.0f, (float)npT);
  float tl = loc / npos;
  float tc = confl / npos;
  int ic = *iouCnt;
  float mean = (ic > 0) ? (*iouSum / (float)(ic > 1 ? ic : 1)) : 0.1f;
  float q = fminf(fmaxf(1.0f - mean, 0.1f), 0.9f);
  float dyn = 1.0f * (1.0f + q);
  out[0] = dyn * tl + 1.0f * tc + 0.2f * cl;
  out[1] = tc;
  out[2] = tl;
  out[3] = cl;
  out[4] = mean;
}

// ---------------------------------------------------------------- launch
extern "C" void kernel_launch(void* const* d_in, const int* in_sizes, int n_in,
                              void* d_out, int out_size, void* d_ws,
                              size_t ws_size, hipStream_t stream) {
  const float* bbox_pred = (const float*)d_in[0];  // [B,A,4]
  const float* conf_pred = (const float*)d_in[1];  // [B,A]
  const float* anchors   = (const float*)d_in[2];  // [A,4]
  const float* gt_boxes  = (const float*)d_in[3];  // [B,G,4]
  float* out = (float*)d_out;

  char* w = (char*)d_ws;
  float* bestIou = (float*)w;                w += (size_t)BI * AI * sizeof(float);
  int* bestIdx = (int*)w;                    w += (size_t)BI * AI * sizeof(int);
  unsigned* negU = (unsigned*)w;             w += (size_t)BI * AI * sizeof(unsigned);
  unsigned long long* bestAnchor = (unsigned long long*)w;
                                             w += (size_t)BI * GI * sizeof(unsigned long long);
  unsigned* hist = (unsigned*)w;             w += (size_t)BI * 256 * sizeof(unsigned);
  int* npP = (int*)w;                        w += (size_t)BI * NB * sizeof(int);
  int* pcP = (int*)w;                        w += (size_t)BI * NB * sizeof(int);
  float* llP = (float*)w;                    w += (size_t)BI * NB * sizeof(float);
  float* pbP = (float*)w;                    w += (size_t)BI * NB * sizeof(float);
  float* isP = (float*)w;                    w += (size_t)BI * NB * sizeof(float);
  int* icP = (int*)w;                        w += (size_t)BI * NB * sizeof(int);
  float* nsP = (float*)w;                    w += (size_t)BI * NB * sizeof(float);
  int* ncP = (int*)w;                        w += (size_t)BI * NB * sizeof(int);
  int* numPos = (int*)w;                     w += BI * sizeof(int);
  int* predCnt = (int*)w;                    w += BI * sizeof(int);
  float* locL = (float*)w;                   w += BI * sizeof(float);
  float* posBce = (float*)w;                 w += BI * sizeof(float);
  int* kNeg = (int*)w;                       w += BI * sizeof(int);
  int* kRem = (int*)w;                       w += BI * sizeof(int);
  unsigned* prefix = (unsigned*)w;           w += BI * sizeof(unsigned);
  float* iouSum = (float*)w;                 w += sizeof(float);
  int* iouCnt = (int*)w;                     w += sizeof(int);

  k_init<<<1, 512, 0, stream>>>(bestAnchor, hist);

  k_iou<<<dim3(AI / 256, BI), 256, 0, stream>>>(
      (const float4*)anchors, (const float4*)gt_boxes, bestIou, bestIdx,
      bestAnchor);

  k_match_p<<<dim3(NB, BI), 256, 0, stream>>>(
      (const float4*)bbox_pred, conf_pred, (const float4*)gt_boxes, bestIou,
      bestIdx, bestAnchor, negU, npP, pcP, llP, pbP, isP, icP);

  k_combine<<<1, 32, 0, stream>>>(npP, pcP, llP, pbP, isP, icP, numPos,
                                  predCnt, locL, posBce, kNeg, kRem, prefix,
                                  iouSum, iouCnt);

  for (int d = 3; d >= 0; --d) {
    k_hist<<<dim3(64, BI), 256, 0, stream>>>(negU, prefix, hist, d);
    k_scan<<<1, 32, 0, stream>>>(hist, prefix, kRem);
  }

  k_negsum_p<<<dim3(NB, BI), 256, 0, stream>>>(negU, prefix, nsP, ncP);

  k_final<<<1, 32, 0, stream>>>(numPos, predCnt, locL, posBce, nsP, ncP,
                                prefix, kNeg, iouSum, iouCnt, out);
  (void)in_sizes; (void)n_in; (void)out_size; (void)ws_size;
}